// SoftKConv_31430570672205
// MI455X (gfx1250) — compile-verified
//
#include <hip/hip_runtime.h>
#include <math.h>

typedef __attribute__((ext_vector_type(2))) float v2f;
typedef __attribute__((ext_vector_type(8))) float v8f;

#define D 128      // feature dim (D_IN == D_OUT)
#define KNB 32     // neighbors per node
#define FSTR 132   // LDS row stride for fk (pad 128 -> 132 to avoid bank conflicts)
#define GSTR 33    // LDS row stride for gram (pad 32 -> 33)
#define NPB 2      // nodes (waves) per block in medoid kernel

__device__ __forceinline__ v8f wmma4f32(v2f a, v2f b, v8f c) {
  // D = A(16x4, f32) * B(4x16, f32) + C(16x16, f32)
  return __builtin_amdgcn_wmma_f32_16x16x4_f32(
      /*neg_a=*/false, a, /*neg_b=*/false, b,
      /*c_mod=*/(short)0, c, /*reuse_a=*/false, /*reuse_b=*/false);
}

// --------------------------------------------------------------------------
// Kernel 1: h = feat @ weight   [N,128] = [N,128] @ [128,128]
// One wave -> one 16x16 output tile; 8 waves/block cover the 128 columns.
// --------------------------------------------------------------------------
__global__ void __launch_bounds__(256)
proj_kernel(const float* __restrict__ feat, const float* __restrict__ weight,
            float* __restrict__ h, int n) {
  const int wave = threadIdx.x >> 5;
  const int lane = threadIdx.x & 31;
  const int r0 = blockIdx.x * 16;
  if (r0 >= n) return;                 // whole-block uniform exit
  const int c0   = wave * 16;
  const int lrow = lane & 15;
  const int kl   = (lane >> 4) * 2;    // lanes 0-15: K+0,K+1 ; lanes 16-31: K+2,K+3
  const int arow = min(r0 + lrow, n - 1);

  v8f acc = {};
  for (int c = 0; c < D / 4; ++c) {
    const int kb = 4 * c + kl;
    // A-frag: feat[row][kb], feat[row][kb+1]  (one 8B load)
    v2f a = *(const v2f*)(feat + (size_t)arow * D + kb);
    // B-frag: weight[kb][col], weight[kb+1][col]
    v2f b;
    b.x = weight[(size_t)kb * D + c0 + lrow];
    b.y = weight[(size_t)(kb + 1) * D + c0 + lrow];
    acc = wmma4f32(a, b, acc);
  }
  // C layout: VGPR i -> M = i + 8*(lane>>4), N = lane&15
  const int mofs = 8 * (lane >> 4);
  for (int i = 0; i < 8; ++i) {
    const int row = r0 + i + mofs;
    if (row < n) h[(size_t)row * D + c0 + lrow] = acc[i];
  }
}

// --------------------------------------------------------------------------
// Kernel 2: per-node soft-medoid. One wave per node.
//   gather fk -> LDS, Gram via 128x v_wmma_f32_16x16x4_f32, distances,
//   softmax + weight correction, out = r^T fk + bias.
// --------------------------------------------------------------------------
__global__ void __launch_bounds__(NPB * 32)
medoid_kernel(const float* __restrict__ h, const float* __restrict__ bias,
              const float* __restrict__ tkw, const int* __restrict__ tki,
              float* __restrict__ out, int n) {
  __shared__ float s_fk[NPB][KNB * FSTR];
  __shared__ float s_gram[NPB][KNB * GSTR];
  __shared__ float s_w[NPB][KNB];
  __shared__ float s_sq[NPB][KNB];
  __shared__ float s_r[NPB][KNB];

  const int wave = threadIdx.x >> 5;
  const int lane = threadIdx.x & 31;
  const int node = blockIdx.x * NPB + wave;
  if (node >= n) return;               // whole-wave uniform exit
  float* fk   = s_fk[wave];
  float* gram = s_gram[wave];

  // ---- gather: lane m pulls neighbor row m (L2-resident) into LDS ----
  const int  nb    = tki[(size_t)node * KNB + lane];
  const bool maskk = (nb < 0);
  const int  row   = maskk ? 0 : nb;
  const float wl   = maskk ? 0.0f : tkw[(size_t)node * KNB + lane];
  s_w[wave][lane] = wl;
  const float4* hrow = (const float4*)(h + (size_t)row * D);
  for (int j = 0; j < D / 4; ++j)
    *(float4*)&fk[lane * FSTR + 4 * j] = hrow[j];
  // same-wave LDS is in-order; no barrier needed (LDS slice is wave-private)

  // ---- Gram = fk @ fk^T via f32 WMMA; B-frag(g) == A-frag(g) by symmetry ----
  const int lrow = lane & 15;
  const int kl   = (lane >> 4) * 2;
  v8f a00 = {}, a01 = {}, a10 = {}, a11 = {};
  for (int c = 0; c < D / 4; ++c) {
    const int kb = 4 * c + kl;
    v2f f0 = *(const v2f*)&fk[lrow * FSTR + kb];         // rows 0-15 fragment
    v2f f1 = *(const v2f*)&fk[(16 + lrow) * FSTR + kb];  // rows 16-31 fragment
    a00 = wmma4f32(f0, f0, a00);
    a01 = wmma4f32(f0, f1, a01);
    a10 = wmma4f32(f1, f0, a10);
    a11 = wmma4f32(f1, f1, a11);
  }
  const int mofs = 8 * (lane >> 4);
  for (int i = 0; i < 8; ++i) {
    gram[(i + mofs) * GSTR + lrow]           = a00[i];
    gram[(i + mofs) * GSTR + 16 + lrow]      = a01[i];
    gram[(16 + i + mofs) * GSTR + lrow]      = a10[i];
    gram[(16 + i + mofs) * GSTR + 16 + lrow] = a11[i];
  }

  // ---- distances + weighted medoid cost; lane == neighbor k ----
  const float sq = gram[lane * GSTR + lane];
  s_sq[wave][lane] = sq;
  float dagg = 0.0f;
  for (int m = 0; m < KNB; ++m) {
    const float g  = gram[lane * GSTR + m];
    const float d2 = sq + s_sq[wave][m] - 2.0f * g;
    const float dist = d2 > 0.0f ? sqrtf(d2) : 0.0f;   // masked m has w==0
    dagg += s_w[wave][m] * dist;
  }
  if (maskk || !isfinite(dagg)) dagg = 3.402823466e38f;

  // ---- softmax(-dagg) with weight correction (wave32 reductions) ----
  float x = -dagg;
  float xm = x;
  for (int o = 16; o > 0; o >>= 1) xm = fmaxf(xm, __shfl_xor(xm, o, 32));
  float e = expf(x - xm);
  float es = e;
  for (int o = 16; o > 0; o >>= 1) es += __shfl_xor(es, o, 32);
  float r = (e / es) * wl;
  float rs = r;
  for (int o = 16; o > 0; o >>= 1) rs += __shfl_xor(rs, o, 32);
  r = maskk ? 0.0f : (r / rs);
  s_r[wave][lane] = r;

  // ---- out[d] = sum_k r[k] * fk[k][d] + bias[d]; lane owns d = 4*lane.. ----
  float4 acc = make_float4(0.f, 0.f, 0.f, 0.f);
  for (int m = 0; m < KNB; ++m) {
    const float rm = s_r[wave][m];
    const float4 f = *(const float4*)&fk[m * FSTR + 4 * lane];
    acc.x += rm * f.x; acc.y += rm * f.y; acc.z += rm * f.z; acc.w += rm * f.w;
  }
  const float4 b = *(const float4*)&bias[4 * lane];
  float4 o4;
  o4.x = acc.x + b.x; o4.y = acc.y + b.y; o4.z = acc.z + b.z; o4.w = acc.w + b.w;
  *(float4*)&out[(size_t)node * D + 4 * lane] = o4;
}

// --------------------------------------------------------------------------
extern "C" void kernel_launch(void* const* d_in, const int* in_sizes, int n_in,
                              void* d_out, int out_size, void* d_ws, size_t ws_size,
                              hipStream_t stream) {
  const float* feat   = (const float*)d_in[0];   // [N,128]
  const float* weight = (const float*)d_in[1];   // [128,128]
  const float* bias   = (const float*)d_in[2];   // [128]
  const float* tkw    = (const float*)d_in[3];   // [N,32]
  const int*   tki    = (const int*)d_in[4];     // [N,32]
  float* out = (float*)d_out;                    // [N,128]
  float* h   = (float*)d_ws;                     // [N,128] scratch (25.6 MB)

  const int N = in_sizes[0] / D;

  const int proj_blocks = (N + 15) / 16;
  proj_kernel<<<proj_blocks, 256, 0, stream>>>(feat, weight, h, N);

  const int med_blocks = (N + NPB - 1) / NPB;
  medoid_kernel<<<med_blocks, NPB * 32, 0, stream>>>(h, bias, tkw, tki, out, N);
}